// GCNLayer_46024869544123
// MI455X (gfx1250) — compile-verified
//
#include <hip/hip_runtime.h>

typedef __attribute__((ext_vector_type(2))) float v2f;
typedef __attribute__((ext_vector_type(8))) float v8f;

#define D_FEAT 64
#define LDS_STRIDE 66   // pad W rows in LDS to avoid bank conflicts

// ---------------------------------------------------------------------------
// Kernel 1: zero the aggregation workspace (float4 stores)
// ---------------------------------------------------------------------------
__global__ __launch_bounds__(256) void gcn_zero_kernel(float4* __restrict__ p, int n4) {
    int i = blockIdx.x * blockDim.x + threadIdx.x;
    if (i < n4) p[i] = make_float4(0.f, 0.f, 0.f, 0.f);
}

// ---------------------------------------------------------------------------
// Kernel 2: edge scatter  agg[row] += val * X[col]
// 16 lanes per edge, each lane covers 4 features (float4 gather, 4 f32 atomics).
// X (25.6 MB) and agg (25.6 MB) are L2-resident on MI455X (192 MB L2), so
// both the gathers and the atomic adds are served by L2, not HBM. The L2
// atomic RMW rate is the roofline limiter for this kernel, so the atomics
// MUST be native global_atomic_add_f32 (unsafeAtomicAdd), never a CAS loop.
// ---------------------------------------------------------------------------
__global__ __launch_bounds__(256) void gcn_scatter_kernel(
    const int*   __restrict__ rows,
    const int*   __restrict__ cols,
    const float* __restrict__ vals,
    const float* __restrict__ X,
    float*       __restrict__ agg,
    int n_edges)
{
    int gid = blockIdx.x * blockDim.x + threadIdx.x;
    int e = gid >> 4;
    if (e >= n_edges) return;
    int f = (gid & 15) << 2;

    int r   = rows[e];
    int c   = cols[e];
    float v = vals[e];

    const float4 x = *(const float4*)(X + (size_t)c * D_FEAT + f);
    float* dst = agg + (size_t)r * D_FEAT + f;

    // hardware fp32 atomic adds (coarse-grained device memory -> safe)
    unsafeAtomicAdd(dst + 0, v * x.x);
    unsafeAtomicAdd(dst + 1, v * x.y);
    unsafeAtomicAdd(dst + 2, v * x.z);
    unsafeAtomicAdd(dst + 3, v * x.w);
}

// ---------------------------------------------------------------------------
// Kernel 3: out = agg @ W.T + b using V_WMMA_F32_16X16X4_F32 (full f32 precision)
//   out[m][n] = sum_k agg[m][k] * W[n][k]   =>  B[k][n] = W[n][k]
// 256 threads = 8 wave32; W staged in LDS once; each wave owns one 16-row tile
// and 4 column tiles (16x64 output), accumulating over K=64 in 16 steps of K=4.
//
// Fragment layouts (ISA 7.12.2, wave32):
//   A 16x4 : lane L -> m = L%16 ; VGPR0/1 = K = (L<16 ? 0,1 : 2,3)  (one float2)
//   B 4x16 : lane L -> n = L%16 ; VGPR0/1 = K = (L<16 ? 0,1 : 2,3)  (mirrored)
//   C/D    : VGPR r -> row (r + (L<16?0:8)), col = L%16
// ---------------------------------------------------------------------------
__global__ __launch_bounds__(256) void gcn_gemm_wmma_kernel(
    const float* __restrict__ agg,
    const float* __restrict__ W,      // [64 x 64], row n = output feature
    const float* __restrict__ bias,   // [64]
    float*       __restrict__ out,
    int n_tiles)                      // number of 16-row tiles
{
    __shared__ float ldsW[D_FEAT * LDS_STRIDE];

    const int tid = threadIdx.x;

    // Cooperative load of W into LDS with padded stride
    for (int i = tid; i < D_FEAT * D_FEAT; i += 256) {
        ldsW[(i >> 6) * LDS_STRIDE + (i & 63)] = W[i];
    }
    __syncthreads();

    const int wave = tid >> 5;
    const int lane = tid & 31;
    const int half = lane >> 4;     // 0: K=0..1 / rows 0..7, 1: K=2..3 / rows 8..15
    const int l16  = lane & 15;
    const int koff = half << 1;

    const int tile = blockIdx.x * 8 + wave;
    if (tile >= n_tiles) return;

    // bias per lane for each of the 4 column tiles
    const float bn0 = bias[ 0 + l16];
    const float bn1 = bias[16 + l16];
    const float bn2 = bias[32 + l16];
    const float bn3 = bias[48 + l16];

    v8f c0 = {}; v8f c1 = {}; v8f c2 = {}; v8f c3 = {};

    const float* arow = agg + ((size_t)tile * 16 + l16) * D_FEAT;

#pragma unroll
    for (int kk = 0; kk < 16; ++kk) {
        const int k0 = kk * 4;

        // A fragment: float2 of agg row (m = tile*16 + l16) at K = k0+koff
        v2f a = *(const v2f*)(arow + k0 + koff);

        // B fragments from LDS: W[n][k0+koff .. +1] for n = nt*16 + l16
        const int kb = k0 + koff;
        v2f b0 = *(const v2f*)(&ldsW[( 0 + l16) * LDS_STRIDE + kb]);
        v2f b1 = *(const v2f*)(&ldsW[(16 + l16) * LDS_STRIDE + kb]);
        v2f b2 = *(const v2f*)(&ldsW[(32 + l16) * LDS_STRIDE + kb]);
        v2f b3 = *(const v2f*)(&ldsW[(48 + l16) * LDS_STRIDE + kb]);

        c0 = __builtin_amdgcn_wmma_f32_16x16x4_f32(false, a, false, b0, (short)0, c0, false, false);
        c1 = __builtin_amdgcn_wmma_f32_16x16x4_f32(false, a, false, b1, (short)0, c1, false, false);
        c2 = __builtin_amdgcn_wmma_f32_16x16x4_f32(false, a, false, b2, (short)0, c2, false, false);
        c3 = __builtin_amdgcn_wmma_f32_16x16x4_f32(false, a, false, b3, (short)0, c3, false, false);
    }

    // Store D + bias. VGPR r holds row (r + half*8), col = l16 within each tile.
    const size_t base = (size_t)tile * 16 * D_FEAT;
#pragma unroll
    for (int r = 0; r < 8; ++r) {
        const size_t m_off = base + (size_t)(r + half * 8) * D_FEAT;
        out[m_off +  0 + l16] = c0[r] + bn0;
        out[m_off + 16 + l16] = c1[r] + bn1;
        out[m_off + 32 + l16] = c2[r] + bn2;
        out[m_off + 48 + l16] = c3[r] + bn3;
    }
}

// ---------------------------------------------------------------------------
extern "C" void kernel_launch(void* const* d_in, const int* in_sizes, int n_in,
                              void* d_out, int out_size, void* d_ws, size_t ws_size,
                              hipStream_t stream) {
    const int*   rows = (const int*)d_in[0];
    const int*   cols = (const int*)d_in[1];
    const float* vals = (const float*)d_in[2];
    const float* X    = (const float*)d_in[3];
    const float* W    = (const float*)d_in[4];
    const float* bias = (const float*)d_in[5];
    float* out = (float*)d_out;
    float* agg = (float*)d_ws;               // n_nodes * 64 floats (25.6 MB)

    const int n_edges = in_sizes[0];
    const int n_nodes = in_sizes[3] / D_FEAT;

    // 1) zero agg
    const int n4 = n_nodes * (D_FEAT / 4);
    gcn_zero_kernel<<<(n4 + 255) / 256, 256, 0, stream>>>((float4*)agg, n4);

    // 2) scatter: 16 threads per edge
    const long long total = (long long)n_edges * 16;
    const int blocks = (int)((total + 255) / 256);
    gcn_scatter_kernel<<<blocks, 256, 0, stream>>>(rows, cols, vals, X, agg, n_edges);

    // 3) WMMA GEMM + bias
    const int n_tiles = (n_nodes + 15) / 16;          // 6250 exact for N=100000
    gcn_gemm_wmma_kernel<<<(n_tiles + 7) / 8, 256, 0, stream>>>(agg, W, bias, out, n_tiles);
}